// GCN_27109833572875
// MI455X (gfx1250) — compile-verified
//
#include <hip/hip_runtime.h>
#include <hip/hip_bf16.h>

typedef float v2f __attribute__((ext_vector_type(2)));
typedef float v8f __attribute__((ext_vector_type(8)));
typedef int   v4i __attribute__((ext_vector_type(4)));

#define NFEAT 512
#define NHID  128
#define NCLS  40

#define GAS __attribute__((address_space(1)))
#define LAS __attribute__((address_space(3)))

#if __has_builtin(__builtin_amdgcn_global_load_async_to_lds_b128) && \
    __has_builtin(__builtin_amdgcn_s_wait_asynccnt)
#define HAVE_ASYNC_LDS 1
#else
#define HAVE_ASYNC_LDS 0
#endif

// Stage one b128 chunk global -> LDS (async-to-LDS if available).
__device__ __forceinline__ void stage_b128(const float* gsrc, float* ldst) {
#if HAVE_ASYNC_LDS
    __builtin_amdgcn_global_load_async_to_lds_b128((GAS v4i*)gsrc, (LAS v4i*)ldst, 0, 0);
#else
    *(float4*)ldst = *(const float4*)gsrc;
#endif
}

__device__ __forceinline__ void stage_fence() {
#if HAVE_ASYNC_LDS
    __builtin_amdgcn_s_wait_asynccnt(0);
#endif
    __syncthreads();
}

// ---------------------------------------------------------------------------
// GEMM1: XW[N,128] = X[N,512] @ W1[512,128]  (fp32 WMMA 16x16x4)
// Block = 256 threads (8 waves). 16-row A tile staged to LDS (async-to-LDS);
// wave w owns output columns [16w, 16w+16).
// ---------------------------------------------------------------------------
__global__ void __launch_bounds__(256) gemm1_wmma(const float* __restrict__ X,
                                                  const float* __restrict__ W1,
                                                  float* __restrict__ XW) {
    constexpr int AS = NFEAT + 4;           // pad: 2064B row stride, 16B aligned
    __shared__ float Asmem[16 * AS];
    const int row0 = blockIdx.x * 16;

    // 16*512 floats = 2048 b128 chunks; 8 per thread
    for (int i = threadIdx.x; i < 16 * (NFEAT / 4); i += 256) {
        int r = i >> 7;                     // /(512/4)
        int c = (i & 127) << 2;
        stage_b128(&X[(size_t)(row0 + r) * NFEAT + c], &Asmem[r * AS + c]);
    }
    stage_fence();

    const int wave = threadIdx.x >> 5;      // 0..7 -> N tile
    const int lane = threadIdx.x & 31;
    const int rl   = lane & 15;             // row (A) / col (B) within tile
    const int kb   = (lane >> 4) << 1;      // half-wave K pair: 0 or 2
    const int col  = wave * 16 + rl;

    v8f acc = {};
    const float* as = &Asmem[rl * AS + kb];
    const float* bp = W1 + (size_t)kb * NHID + col;

    #pragma unroll 4
    for (int k = 0; k < NFEAT; k += 4) {
        v2f a, b;
        a.x = as[k];
        a.y = as[k + 1];
        b.x = bp[(size_t)k * NHID];
        b.y = bp[(size_t)(k + 1) * NHID];
        acc = __builtin_amdgcn_wmma_f32_16x16x4_f32(false, a, false, b,
                                                    (short)0, acc, false, false);
    }

    // C/D layout: VGPR i -> M = i (lanes 0-15) or i+8 (lanes 16-31), N = rl
    const int ro = (lane < 16) ? 0 : 8;
    #pragma unroll
    for (int i = 0; i < 8; ++i)
        XW[(size_t)(row0 + ro + i) * NHID + col] = acc[i];
}

// ---------------------------------------------------------------------------
// GEMM2: HW[N,40] = relu(H)[N,128] @ W2[128,40]  (fp32 WMMA, N padded to 48)
// H already contains (spmm + b1); ReLU folded into the A-fragment build.
// Block = 96 threads (3 waves). 16-row A tile staged to LDS.
// ---------------------------------------------------------------------------
__global__ void __launch_bounds__(96) gemm2_wmma(const float* __restrict__ H,
                                                 const float* __restrict__ W2,
                                                 float* __restrict__ HW) {
    constexpr int AS = NHID + 4;            // 528B row stride, 16B aligned
    __shared__ float Asmem[16 * AS];
    const int row0 = blockIdx.x * 16;

    // 16*128 floats = 512 b128 chunks
    for (int i = threadIdx.x; i < 16 * (NHID / 4); i += 96) {
        int r = i >> 5;                     // /(128/4)
        int c = (i & 31) << 2;
        stage_b128(&H[(size_t)(row0 + r) * NHID + c], &Asmem[r * AS + c]);
    }
    stage_fence();

    const int wave = threadIdx.x >> 5;      // 0..2
    const int lane = threadIdx.x & 31;
    const int rl   = lane & 15;
    const int kb   = (lane >> 4) << 1;
    const int col  = wave * 16 + rl;        // 0..47
    const bool cv  = (col < NCLS);

    v8f acc = {};
    const float* as = &Asmem[rl * AS + kb];

    #pragma unroll 4
    for (int k = 0; k < NHID; k += 4) {
        v2f a, b;
        a.x = fmaxf(as[k], 0.0f);           // ReLU folded here
        a.y = fmaxf(as[k + 1], 0.0f);
        b.x = cv ? W2[(size_t)(k + kb) * NCLS + col] : 0.0f;
        b.y = cv ? W2[(size_t)(k + kb + 1) * NCLS + col] : 0.0f;
        acc = __builtin_amdgcn_wmma_f32_16x16x4_f32(false, a, false, b,
                                                    (short)0, acc, false, false);
    }

    if (cv) {
        const int ro = (lane < 16) ? 0 : 8;
        #pragma unroll
        for (int i = 0; i < 8; ++i)
            HW[(size_t)(row0 + ro + i) * NCLS + col] = acc[i];
    }
}

// ---------------------------------------------------------------------------
// H[i] = b1[i % 128]  (seed accumulator with bias; replaces memset + bias pass)
// ---------------------------------------------------------------------------
__global__ void __launch_bounds__(256) bias_init(float* __restrict__ H,
                                                 const float* __restrict__ b1, int n) {
    int i = blockIdx.x * 256 + threadIdx.x;
    if (i < n) H[i] = b1[i & (NHID - 1)];
}

// ---------------------------------------------------------------------------
// SpMM layer 1: H[dst] += w * XW[src], 128 feats. One wave per edge,
// each lane handles a float4.
// ---------------------------------------------------------------------------
__global__ void __launch_bounds__(256) spmm1_scatter(const int* __restrict__ ei,
                                                     const float* __restrict__ ew,
                                                     const float* __restrict__ XW,
                                                     float* __restrict__ H, int E) {
    long long gid = (long long)blockIdx.x * 256 + threadIdx.x;
    int e = (int)(gid >> 5);
    if (e >= E) return;
    int lane = threadIdx.x & 31;

    int   src = ei[e];
    int   dst = ei[(size_t)E + e];
    float w   = ew[e];

    float4 v = ((const float4*)(XW + (size_t)src * NHID))[lane];
    float* hp = H + (size_t)dst * NHID + lane * 4;
    atomicAdd(hp + 0, w * v.x);
    atomicAdd(hp + 1, w * v.y);
    atomicAdd(hp + 2, w * v.z);
    atomicAdd(hp + 3, w * v.w);
}

// ---------------------------------------------------------------------------
// SpMM layer 2: O2[dst] += w * HW[src], 40 feats. 10 threads x float4 per edge.
// ---------------------------------------------------------------------------
__global__ void __launch_bounds__(256) spmm2_scatter(const int* __restrict__ ei,
                                                     const float* __restrict__ ew,
                                                     const float* __restrict__ HW,
                                                     float* __restrict__ O2, int E) {
    long long gid = (long long)blockIdx.x * 256 + threadIdx.x;
    long long total = (long long)E * 10;
    if (gid >= total) return;
    int e = (int)(gid / 10);
    int q = (int)(gid % 10);

    int   src = ei[e];
    int   dst = ei[(size_t)E + e];
    float w   = ew[e];

    float4 v = ((const float4*)(HW + (size_t)src * NCLS))[q];
    float* op = O2 + (size_t)dst * NCLS + q * 4;
    atomicAdd(op + 0, w * v.x);
    atomicAdd(op + 1, w * v.y);
    atomicAdd(op + 2, w * v.z);
    atomicAdd(op + 3, w * v.w);
}

// ---------------------------------------------------------------------------
// out[row] = log_softmax(O2[row] + b2), 40 classes; one wave32 per row.
// ---------------------------------------------------------------------------
__global__ void __launch_bounds__(256) logsoftmax_rows(const float* __restrict__ O2,
                                                       const float* __restrict__ b2,
                                                       float* __restrict__ out, int nrows) {
    int gid  = blockIdx.x * 256 + threadIdx.x;
    int row  = gid >> 5;
    int lane = threadIdx.x & 31;
    if (row >= nrows) return;

    const float* r = O2 + (size_t)row * NCLS;
    float x0 = r[lane] + b2[lane];
    float x1 = (lane < 8) ? (r[32 + lane] + b2[32 + lane]) : -3.0e38f;

    float m = fmaxf(x0, x1);
    #pragma unroll
    for (int o = 16; o >= 1; o >>= 1)
        m = fmaxf(m, __shfl_xor(m, o, 32));

    float s = expf(x0 - m) + ((lane < 8) ? expf(x1 - m) : 0.0f);
    #pragma unroll
    for (int o = 16; o >= 1; o >>= 1)
        s += __shfl_xor(s, o, 32);

    float ls = logf(s);
    float* o_ = out + (size_t)row * NCLS;
    o_[lane] = x0 - m - ls;
    if (lane < 8) o_[32 + lane] = x1 - m - ls;
}

// ---------------------------------------------------------------------------
extern "C" void kernel_launch(void* const* d_in, const int* in_sizes, int n_in,
                              void* d_out, int out_size, void* d_ws, size_t ws_size,
                              hipStream_t stream) {
    const float* X  = (const float*)d_in[0];   // [N, 512]
    const float* ew = (const float*)d_in[1];   // [E]
    const float* W1 = (const float*)d_in[2];   // [512, 128]
    const float* b1 = (const float*)d_in[3];   // [128]
    const float* W2 = (const float*)d_in[4];   // [128, 40]
    const float* b2 = (const float*)d_in[5];   // [40]
    const int*   ei = (const int*)d_in[6];     // [2, E]
    float* out = (float*)d_out;                // [N, 40]

    const int N = in_sizes[0] / NFEAT;
    const int E = in_sizes[1];

    // workspace layout (256B aligned):
    //   buf0: XW [N,128]  (reused as HW [N,40] after SpMM1 consumes it)
    //   buf1: H  [N,128]
    //   buf2: O2 [N,40]
    char* ws = (char*)d_ws;
    size_t szH = ((size_t)N * NHID * sizeof(float) + 255) & ~(size_t)255;
    float* XW = (float*)(ws);
    float* H  = (float*)(ws + szH);
    float* O2 = (float*)(ws + 2 * szH);
    float* HW = XW;
    (void)ws_size; (void)n_in; (void)out_size;

    const unsigned mtiles = (unsigned)(N / 16);              // N = 50000 = 16*3125

    // layer 1:  H = spmm(XW1) + b1  (bias seeded into accumulator)
    gemm1_wmma<<<mtiles, 256, 0, stream>>>(X, W1, XW);
    {
        int n = N * NHID;
        bias_init<<<(unsigned)((n + 255) / 256), 256, 0, stream>>>(H, b1, n);
    }
    {
        long long threads = (long long)E * 32;
        unsigned blocks = (unsigned)((threads + 255) / 256);
        spmm1_scatter<<<blocks, 256, 0, stream>>>(ei, ew, XW, H, E);
    }

    // layer 2:  out = log_softmax(spmm(relu(H) @ W2) + b2)
    gemm2_wmma<<<mtiles, 96, 0, stream>>>(H, W2, HW);
    (void)hipMemsetAsync(O2, 0, (size_t)N * NCLS * sizeof(float), stream);
    {
        long long threads = (long long)E * 10;
        unsigned blocks = (unsigned)((threads + 255) / 256);
        spmm2_scatter<<<blocks, 256, 0, stream>>>(ei, ew, HW, O2, E);
    }
    {
        long long threads = (long long)N * 32;
        unsigned blocks = (unsigned)((threads + 255) / 256);
        logsoftmax_rows<<<blocks, 256, 0, stream>>>(O2, b2, out, N);
    }
}